// DfOpCoefLoop_30185030156317
// MI455X (gfx1250) — compile-verified
//
#include <hip/hip_runtime.h>
#include <cstdint>

// Problem constants (from the reference)
#define DF_ORDER   5
#define DF_BINS    96
#define FREQ_BINS  481
#define TAIL_BINS  (FREQ_BINS - DF_BINS)   // 385
#define BATCH      32
#define TSTEPS     1000
#define TILE_T     8                        // time steps per block (1000/8 = 125 tiles)
#define TILE_ROWS  (TILE_T + DF_ORDER - 1)  // 12 staged spec rows (t0-2 .. t0+TILE_T+1)
#define NTHREADS   256

#define AS_GLOBAL __attribute__((address_space(1)))
#define AS_LDS    __attribute__((address_space(3)))

// Matches the builtin's parameter type (probed via hipcc diagnostic):
// 'int __attribute__((vector_size(8))) __device__ *'
typedef int   v2i_t __attribute__((vector_size(8)));
typedef float f2v_t __attribute__((ext_vector_type(2)));

// CDNA5 async global->LDS path (ASYNCcnt). Guarded so the host pass compiles.
#if defined(__HIP_DEVICE_COMPILE__) && __has_builtin(__builtin_amdgcn_global_load_async_to_lds_b64)
#define USE_ASYNC_LDS 1
#else
#define USE_ASYNC_LDS 0
#endif

// Non-temporal (TH=NT) 8-byte load/store helpers — all global streams here are
// touch-once, so keep them out of WGP$/L2 residency.
__device__ __forceinline__ float2 nt_load2(const float2* p) {
    f2v_t v = __builtin_nontemporal_load((const f2v_t*)p);
    return make_float2(v.x, v.y);
}
__device__ __forceinline__ void nt_store2(float2* p, float2 d) {
    f2v_t v = {d.x, d.y};
    __builtin_nontemporal_store(v, (f2v_t*)p);
}

__global__ __launch_bounds__(NTHREADS)
void df_coef_loop_kernel(const float* __restrict__ spec,
                         const float* __restrict__ coefs,
                         const float* __restrict__ alpha,
                         float* __restrict__ out)
{
    const int tid = threadIdx.x;
    const int b   = blockIdx.y;
    const int t0  = blockIdx.x * TILE_T;

    // 12 rows x 96 complex = 9216 B of LDS
    __shared__ float2 s_x[TILE_ROWS * DF_BINS];

    const float2* __restrict__ spec2 = (const float2*)spec;   // (b*T+t)*481 + f
    const float2* __restrict__ coef2 = (const float2*)coefs;  // ((b*T+t)*5+i)*96 + f
    float2* __restrict__ out2        = (float2*)out;

    // ---- Stage spec[b, t0-2 .. t0+TILE_T+1, 0:96] into LDS (async DMA path) ----
    for (int idx = tid; idx < TILE_ROWS * DF_BINS; idx += NTHREADS) {
        const int r = idx / DF_BINS;
        const int f = idx - r * DF_BINS;
        const int t = t0 - 2 + r;
        if (t >= 0 && t < TSTEPS) {
            const float2* g = spec2 + ((b * TSTEPS + t) * FREQ_BINS + f);
#if USE_ASYNC_LDS
            __builtin_amdgcn_global_load_async_to_lds_b64(
                (AS_GLOBAL v2i_t*)g, (AS_LDS v2i_t*)&s_x[idx],
                /*offset=*/0, /*cpol=*/0);
#else
            s_x[idx] = *g;
#endif
        } else {
            s_x[idx] = make_float2(0.f, 0.f);  // zero-pad outside [0, T)
        }
    }
#if USE_ASYNC_LDS
#if __has_builtin(__builtin_amdgcn_s_wait_asynccnt)
    __builtin_amdgcn_s_wait_asynccnt(0);
#else
    asm volatile("s_wait_asynccnt 0" ::: "memory");
#endif
#endif
    __syncthreads();

    // ---- 5-tap complex FIR + alpha blend on bins [0,96) ----
    // 8*96 = 768 outputs = exactly 3 iterations of 256 threads.
    for (int idx = tid; idx < TILE_T * DF_BINS; idx += NTHREADS) {
        const int tl = idx / DF_BINS;
        const int f  = idx - tl * DF_BINS;
        const int t  = t0 + tl;
        const int bt = b * TSTEPS + t;
        const int cbase = (bt * DF_ORDER) * DF_BINS + f;

        float re = 0.f, im = 0.f;
#pragma unroll
        for (int i = 0; i < DF_ORDER; ++i) {
            const float2 w = s_x[(tl + i) * DF_BINS + f];        // x[t + i - 2, f]
            const float2 c = nt_load2(coef2 + cbase + i * DF_BINS); // coefs[b,t,i,f]
            re += w.x * c.x - w.y * c.y;
            im += w.y * c.x + w.x * c.y;
        }
        const float  a = alpha[bt];
        const float2 s = s_x[(tl + 2) * DF_BINS + f];            // dry spec at time t
        float2 o;
        o.x = re * a + s.x * (1.f - a);
        o.y = im * a + s.y * (1.f - a);
        nt_store2(out2 + bt * FREQ_BINS + f, o);
    }

    // ---- Pass-through copy of bins [96, 481), fully streaming ----
    for (int idx = tid; idx < TILE_T * TAIL_BINS; idx += NTHREADS) {
        const int tl = idx / TAIL_BINS;
        const int f  = DF_BINS + (idx - tl * TAIL_BINS);
        const int o  = (b * TSTEPS + t0 + tl) * FREQ_BINS + f;
        nt_store2(out2 + o, nt_load2(spec2 + o));
    }
}

extern "C" void kernel_launch(void* const* d_in, const int* in_sizes, int n_in,
                              void* d_out, int out_size, void* d_ws, size_t ws_size,
                              hipStream_t stream) {
    (void)in_sizes; (void)n_in; (void)out_size; (void)d_ws; (void)ws_size;
    const float* spec  = (const float*)d_in[0];
    const float* coefs = (const float*)d_in[1];
    const float* alpha = (const float*)d_in[2];
    float* out = (float*)d_out;

    dim3 grid(TSTEPS / TILE_T, BATCH);   // (125, 32) = 4000 blocks
    df_coef_loop_kernel<<<grid, NTHREADS, 0, stream>>>(spec, coefs, alpha, out);
}